// PositionalEncoding_31559419691757
// MI455X (gfx1250) — compile-verified
//
#include <hip/hip_runtime.h>
#include <hip/hip_bf16.h>

typedef float v2f __attribute__((ext_vector_type(2)));
typedef float v8f __attribute__((ext_vector_type(8)));

#define D_MODEL 1024
#define NT 4            // position tiles (16 rows each) per wave
#define WAVES_PER_BLOCK 8

// out[pos*1024 + 2i]   = sin(pos * 10000^(-2i/1024))
// out[pos*1024 + 2i+1] = cos(pos * 10000^(-2i/1024))
//
// angles = pos (x) inv_freq  -> rank-1 outer product done on the WMMA pipe
// in 16x16 tiles; sin/cos via hardware v_sin/v_cos on revolutions.
__global__ __launch_bounds__(256) void pe_wmma_kernel(float* __restrict__ out,
                                                      int seq_len) {
  const int lane = threadIdx.x & 31;
  const int wave = threadIdx.x >> 5;
  const int wglobal = blockIdx.x * WAVES_PER_BLOCK + wave;

  const int tile_n = wglobal & 31;   // 32 tiles of 16 pair-indices = 512 pairs
  const int pgroup = wglobal >> 5;   // position group: NT*16 rows
  const int i0 = tile_n * 16;
  const int pos_base = pgroup * (16 * NT);

  const bool lo = lane < 16;
  const int l15 = lane & 15;

  // rfreq_n = 10000^(-2*(i0+n)/1024) * 1/(2*pi)   (revolutions per position)
  const float LOG2_10000 = 13.28771237954945f;
  const float INV_2PI    = 0.15915493667125702f;
  float e = -(LOG2_10000 * 2.0f / (float)D_MODEL) * (float)(i0 + l15);
  float rf = __builtin_amdgcn_exp2f(e) * INV_2PI;   // v_exp_f32, hoisted

  // B matrix 4x16 f32: K=0 row = rfreq, K=1..3 = 0. Lanes 0-15 hold K=0,1.
  v2f b;
  b.x = lo ? rf : 0.0f;
  b.y = 0.0f;

  for (int t = 0; t < NT; ++t) {
    const int pos0 = pos_base + t * 16;

    // A matrix 16x4 f32: column K=0 = positions, K=1..3 = 0.
    v2f a;
    a.x = lo ? (float)(pos0 + l15) : 0.0f;
    a.y = 0.0f;

    v8f c = {};
    // D[m][n] = pos_m * rfreq_n  (revolutions) on the matrix pipe
    v8f d = __builtin_amdgcn_wmma_f32_16x16x4_f32(
        /*neg_a=*/false, a, /*neg_b=*/false, b,
        /*c_mod=*/(short)0, c, /*reuse_a=*/false, /*reuse_b=*/false);

    // D layout: VGPR r, lane l -> M = r + (l>=16 ? 8 : 0), N = l&15
    const int m_base = pos0 + (lo ? 0 : 8);
    float* p = out + (size_t)m_base * D_MODEL + 2 * (i0 + l15);

#pragma unroll
    for (int r = 0; r < 8; ++r) {
      float rev = d[r];
      float fr = __builtin_amdgcn_fractf(rev);   // exact range reduction
      v2f sc;
      sc.x = __builtin_amdgcn_sinf(fr);          // v_sin_f32 (unit = 2*pi)
      sc.y = __builtin_amdgcn_cosf(fr);          // v_cos_f32
      // lanes 0-15: 16 consecutive pairs -> 128B contiguous segment
      *(v2f*)(p + (size_t)r * D_MODEL) = sc;     // global_store_b64
    }
  }
}

extern "C" void kernel_launch(void* const* d_in, const int* in_sizes, int n_in,
                              void* d_out, int out_size, void* d_ws, size_t ws_size,
                              hipStream_t stream) {
  (void)d_in; (void)in_sizes; (void)n_in; (void)d_ws; (void)ws_size;
  float* out = (float*)d_out;

  const int seq_len = out_size / D_MODEL;          // 8192
  const int pos_tiles = seq_len / 16;              // 512
  const int pgroups = pos_tiles / NT;              // 128
  const int total_waves = 32 * pgroups;            // 4096
  const int blocks = total_waves / WAVES_PER_BLOCK; // 512

  pe_wmma_kernel<<<blocks, WAVES_PER_BLOCK * 32, 0, stream>>>(out, seq_len);
}